// TransducerLoss_68461778698900
// MI455X (gfx1250) — compile-verified
//
#include <hip/hip_runtime.h>
#include <math.h>

// ---------------------------------------------------------------------------
// Transducer forward loss for MI455X (gfx1250, wave32).
//
// Phase 1 (bandwidth bound, ~500 MB read @ 23.3 TB/s => ~21 us floor):
//   per (b,t,u) row of A=512 logits compute lse = logsumexp, then
//   emit[b,t,u]  = logits[...,label[b,u]] - lse   (u < L)
//   blank[b,t,u] = logits[...,blank]      - lse
//   One wave per row; non-temporal float4 loads (row is read exactly once);
//   fp32 WMMA (16x16x4) chain does the cross-lane exp-sum on the matrix pipe;
//   the two gathered logits come from registers via __shfl (no re-loads).
//
// Phase 2 (tiny DP): anti-diagonal wavefront of the alpha recursion, one
//   128-thread block per batch, one barrier per diagonal (400 diagonals);
//   emit/blank (~2 MB) stay resident in the 192 MB L2.
//
// Phase 3: loss_b = U*sum(g^2) - (sum g)^2 with g = log_r - 2*s  (closed form
//   of the masked pairwise sum), then mean over batches.
// ---------------------------------------------------------------------------

#define NEGF (-1e30f)

typedef float v2f __attribute__((ext_vector_type(2)));
typedef float v4f __attribute__((ext_vector_type(4)));
typedef float v8f __attribute__((ext_vector_type(8)));

__device__ __forceinline__ float logaddexpf_(float a, float b) {
    float m = fmaxf(a, b);
    float n = fminf(a, b);
    return m + log1pf(__expf(n - m));
}

// Full-wave (32 lane) sum of two partials per lane (64 fp32 values total).
// Stage 1: D1 = A(16x4 partials) x ones => 16 row sums q_0..q_15 (D1 vgpr j
// holds q_j in lanes 0-15, q_{j+8} in lanes 16-31).  Stage 2: four
// ones x (pairs of D1 vgprs) WMMAs; each output element equals the sum of the
// 4 q's in its column, so 3 adds broadcast the grand total to all lanes.
// fp32 MACs throughout, deterministic.
__device__ __forceinline__ float wave_sum64(float p0, float p1) {
#if defined(__gfx1250__) && __has_builtin(__builtin_amdgcn_wmma_f32_16x16x4_f32)
    v2f a;   a[0] = p0;     a[1] = p1;
    v2f one; one[0] = 1.0f; one[1] = 1.0f;
    v8f z = {0.f, 0.f, 0.f, 0.f, 0.f, 0.f, 0.f, 0.f};
    v8f d1 = __builtin_amdgcn_wmma_f32_16x16x4_f32(false, a, false, one,
                                                   (short)0, z, false, false);
    v2f b01; b01[0] = d1[0]; b01[1] = d1[1];
    v2f b23; b23[0] = d1[2]; b23[1] = d1[3];
    v2f b45; b45[0] = d1[4]; b45[1] = d1[5];
    v2f b67; b67[0] = d1[6]; b67[1] = d1[7];
    v8f s01 = __builtin_amdgcn_wmma_f32_16x16x4_f32(false, one, false, b01,
                                                    (short)0, z, false, false);
    v8f s23 = __builtin_amdgcn_wmma_f32_16x16x4_f32(false, one, false, b23,
                                                    (short)0, z, false, false);
    v8f s45 = __builtin_amdgcn_wmma_f32_16x16x4_f32(false, one, false, b45,
                                                    (short)0, z, false, false);
    v8f s67 = __builtin_amdgcn_wmma_f32_16x16x4_f32(false, one, false, b67,
                                                    (short)0, z, false, false);
    return s01[0] + s23[0] + s45[0] + s67[0];
#else
    float s = p0 + p1;
#pragma unroll
    for (int off = 16; off; off >>= 1) s += __shfl_xor(s, off, 32);
    return s;
#endif
}

// Extract element e (0..511, wave-uniform) of the row held in registers:
// lane l owns elements {4l..4l+3, 128+4l.., 256+4l.., 384+4l..} in v0..v3.
// Uniform selects + one shuffle; avoids re-reading (possibly NT-evicted) data.
__device__ __forceinline__ float row_extract(v4f v0, v4f v1, v4f v2, v4f v3,
                                             int e) {
    v4f c = (e < 256) ? ((e < 128) ? v0 : v1) : ((e < 384) ? v2 : v3);
    int comp = e & 3;
    float x = (comp == 0) ? c[0] : (comp == 1) ? c[1] : (comp == 2) ? c[2] : c[3];
    return __shfl(x, (e >> 2) & 31, 32);
}

// ---------------------------------------------------------------------------
// Kernel 1: streaming logsumexp + emit/blank extraction.  A is fixed at 512:
// each lane holds 16 values (4x float4, fully coalesced 512B/wave/load).
// ---------------------------------------------------------------------------
__global__ __launch_bounds__(256) void k_lse_emit_blank(
    const float* __restrict__ logits, const int* __restrict__ labels,
    const int* __restrict__ blankArr, float* __restrict__ emitOut,
    float* __restrict__ blankOut, int Tm, int Um, int L, int A,
    int totalRows) {
    const int lane = threadIdx.x & 31;
    const int waveInBlock = threadIdx.x >> 5;
    const int wavesPerBlock = blockDim.x >> 5;
    const int wave = blockIdx.x * wavesPerBlock + waveInBlock;
    const int nWaves = gridDim.x * wavesPerBlock;
    const int blankIdx = blankArr[0];

    for (int row = wave; row < totalRows; row += nWaves) {
        const float* __restrict__ rp = logits + (size_t)row * A;

        // Prefetch this wave's next row (global_prefetch_b8 on gfx1250).
        int nrow = row + nWaves;
        if (nrow < totalRows) {
            __builtin_prefetch(logits + (size_t)nrow * A + lane * 16, 0, 1);
        }

        // Non-temporal: each logits row is consumed exactly once; keep the
        // 496 MB stream from evicting the emit/blank working set in L2.
        const v4f* rp4 = (const v4f*)rp;
        v4f v0 = __builtin_nontemporal_load(rp4 + lane);
        v4f v1 = __builtin_nontemporal_load(rp4 + lane + 32);
        v4f v2 = __builtin_nontemporal_load(rp4 + lane + 64);
        v4f v3 = __builtin_nontemporal_load(rp4 + lane + 96);

        // Per-lane max of 16 values, then wave max via xor shuffles.
        float m = fmaxf(fmaxf(fmaxf(v0[0], v0[1]), fmaxf(v0[2], v0[3])),
                        fmaxf(fmaxf(v1[0], v1[1]), fmaxf(v1[2], v1[3])));
        m = fmaxf(m, fmaxf(fmaxf(fmaxf(v2[0], v2[1]), fmaxf(v2[2], v2[3])),
                           fmaxf(fmaxf(v3[0], v3[1]), fmaxf(v3[2], v3[3]))));
#pragma unroll
        for (int off = 16; off; off >>= 1) m = fmaxf(m, __shfl_xor(m, off, 32));

        // Two fp32 partials per lane (8 exps each) -> 64 values for WMMA.
        float p0 = __expf(v0[0] - m) + __expf(v0[1] - m) + __expf(v0[2] - m) +
                   __expf(v0[3] - m) + __expf(v1[0] - m) + __expf(v1[1] - m) +
                   __expf(v1[2] - m) + __expf(v1[3] - m);
        float p1 = __expf(v2[0] - m) + __expf(v2[1] - m) + __expf(v2[2] - m) +
                   __expf(v2[3] - m) + __expf(v3[0] - m) + __expf(v3[1] - m) +
                   __expf(v3[2] - m) + __expf(v3[3] - m);

        float S = wave_sum64(p0, p1);   // EXEC all-ones (uniform control flow)
        float lse = m + __logf(S);

        // Wave-uniform (b, t, u) decode, done with full EXEC.
        int b = row / (Tm * Um);
        int rem = row - b * (Tm * Um);
        int t = rem / Um;
        int u = rem - t * Um;

        // Gather the two needed logits from registers (no memory re-reads).
        float blankVal = row_extract(v0, v1, v2, v3, blankIdx) - lse;
        float emitVal = 0.0f;
        if (u < L) {
            int lab = labels[b * L + u];  // wave-uniform -> scalar load
            emitVal = row_extract(v0, v1, v2, v3, lab) - lse;
        }

        if (lane == 0) {
            blankOut[row] = blankVal;     // blank layout == row order
            if (u < L) {
                emitOut[(size_t)(b * Tm + t) * L + u] = emitVal;
            }
        }
    }
}

// ---------------------------------------------------------------------------
// Kernel 2: anti-diagonal wavefront alpha recursion + closed-form loss.
// One block per batch, 128 threads (u = threadIdx.x), 1 barrier per diagonal.
// ---------------------------------------------------------------------------
__global__ __launch_bounds__(128) void k_alpha_loss(
    const float* __restrict__ emitIn, const float* __restrict__ blankIn,
    const float* __restrict__ log_r, const int* __restrict__ Tarr,
    const int* __restrict__ Uarr, float* __restrict__ lossb, int Tm, int Um,
    int L, float invB) {
    __shared__ float bufA[128];
    __shared__ float bufB[128];
    __shared__ float alast[128];
    __shared__ float red[128];

    const int b = blockIdx.x;
    const int u = threadIdx.x;
    const int Tb = Tarr[b];
    const int Ub = Uarr[b];
    const float* __restrict__ bl = blankIn + (size_t)b * Tm * Um;
    const float* __restrict__ em = emitIn + (size_t)b * Tm * L;

    float* prev = bufA;
    float* cur = bufB;
    const int nd = Tb + Um - 1;  // number of anti-diagonals

    for (int d = 0; d < nd; ++d) {
        int t = d - u;
        float val = NEGF;
        if (u < Um && t >= 0 && t < Tb) {
            if (t == 0 && u == 0) {
                val = 0.0f;
            } else {
                float ta = (t > 0) ? prev[u] + bl[(t - 1) * Um + u] : NEGF;
                float tb = (u > 0) ? prev[u - 1] + em[t * L + (u - 1)] : NEGF;
                val = logaddexpf_(ta, tb);
            }
            if (t == Tb - 1) alast[u] = val;  // row t = T-1 as it appears
        }
        cur[u] = val;
        __syncthreads();
        float* tmp = prev; prev = cur; cur = tmp;
    }

    // g_k = log_r[k] - 2*s[k],  s[k] = (alpha[T-1][k] + blank[T-1][k]) / T
    float g = 0.0f, g2 = 0.0f;
    if (u < Ub) {
        float s = (alast[u] + bl[(Tb - 1) * Um + u]) / (float)Tb;
        float gg = log_r[b * L + u] - 2.0f * s;
        g = gg;
        g2 = gg * gg;
    }

    red[u] = g;
    __syncthreads();
#pragma unroll
    for (int off = 64; off; off >>= 1) {
        if (u < off) red[u] += red[u + off];
        __syncthreads();
    }
    float sumg = red[0];
    __syncthreads();
    red[u] = g2;
    __syncthreads();
#pragma unroll
    for (int off = 64; off; off >>= 1) {
        if (u < off) red[u] += red[u + off];
        __syncthreads();
    }
    if (u == 0) {
        float sumg2 = red[0];
        // sum_{i<j<U} (g_i - g_j)^2 = U * sum(g^2) - (sum g)^2
        float loss = (float)Ub * sumg2 - sumg * sumg;
        lossb[b] = loss * invB;
    }
}

// ---------------------------------------------------------------------------
// Kernel 3: deterministic final sum over batches.
// ---------------------------------------------------------------------------
__global__ void k_finalize(const float* __restrict__ lossb,
                           float* __restrict__ out, int B) {
    if (threadIdx.x == 0 && blockIdx.x == 0) {
        float s = 0.0f;
        for (int i = 0; i < B; ++i) s += lossb[i];
        out[0] = s;
    }
}

extern "C" void kernel_launch(void* const* d_in, const int* in_sizes, int n_in,
                              void* d_out, int out_size, void* d_ws,
                              size_t ws_size, hipStream_t stream) {
    const float* logits = (const float*)d_in[0];
    const int* labels = (const int*)d_in[1];
    const float* log_r = (const float*)d_in[2];
    const int* Tarr = (const int*)d_in[3];
    const int* Uarr = (const int*)d_in[4];
    const int* blankArr = (const int*)d_in[5];

    const int B = in_sizes[3];            // 8
    const int L = in_sizes[1] / B;        // 100
    const int Um = L + 1;                 // 101
    const int A = 512;                    // vocab (fixed by reference setup)
    const int Tm = (int)((long long)in_sizes[0] / ((long long)B * Um * A));
    const int totalRows = B * Tm * Um;    // 242400

    // Workspace layout (all written before read; harness poison is fine):
    float* w_emit = (float*)d_ws;                        // B*Tm*L floats
    float* w_blank = w_emit + (size_t)B * Tm * L;        // B*Tm*Um floats
    float* w_loss = w_blank + (size_t)B * Tm * Um;       // B floats

    // Phase 1: one wave per row, grid-stride.
    const int threads = 256;
    const int wavesPerBlock = threads / 32;
    int blocks = (totalRows + wavesPerBlock - 1) / wavesPerBlock;
    if (blocks > 4096) blocks = 4096;
    k_lse_emit_blank<<<blocks, threads, 0, stream>>>(
        logits, labels, blankArr, w_emit, w_blank, Tm, Um, L, A, totalRows);

    // Phase 2: one block per batch.
    k_alpha_loss<<<B, 128, 0, stream>>>(w_emit, w_blank, log_r, Tarr, Uarr,
                                        w_loss, Tm, Um, L, 1.0f / (float)B);

    // Phase 3: scalar output.
    k_finalize<<<1, 32, 0, stream>>>(w_loss, (float*)d_out, B);
}